// CausalConv1d_10900626997551
// MI455X (gfx1250) — compile-verified
//
#include <hip/hip_runtime.h>
#include <hip/hip_bf16.h>

// Problem constants (match the JAX reference).
constexpr int Bc = 4;
constexpr int Sc = 4096;
constexpr int Dc = 2048;
constexpr int Kc = 7;

constexpr int T    = 256;  // threads per block (8 wave32s)
constexpr int VEC  = 4;    // channels per thread (float4)
constexpr int TS   = 64;   // timesteps per block
constexpr int NBUF = 4;    // async loads in flight
constexpr int RING = 5;    // LDS ring slots (> NBUF: no reuse race)

static_assert((TS - NBUF) % RING == 0, "steady loop must unroll by RING");

#if __has_builtin(__builtin_amdgcn_global_load_async_to_lds_b128) && \
    __has_builtin(__builtin_amdgcn_s_wait_asynccnt)
#define CC_ASYNC 1
#else
#define CC_ASYNC 0
#endif

#if CC_ASYNC
// Builtin signature (from hipcc diagnostic): src is v4i in addrspace(1),
// dst is v4i in addrspace(3). 16 bytes per lane, tracked by ASYNCcnt.
typedef int v4i __attribute__((vector_size(16)));
typedef __attribute__((address_space(1))) v4i* g128p;
typedef __attribute__((address_space(3))) v4i* l128p;

__device__ __forceinline__ void async_cp16(const void* g, void* l) {
  __builtin_amdgcn_global_load_async_to_lds_b128((g128p)(void*)g, (l128p)l,
                                                 /*offset=*/0, /*cpol=*/0);
}
#endif

__global__ __launch_bounds__(T)
void causal_conv1d_fwd(const float* __restrict__ x,
                       const float* __restrict__ cs,
                       const float* __restrict__ Wt,
                       const float* __restrict__ bias,
                       float* __restrict__ out) {
  const int b  = blockIdx.z;
  const int d0 = blockIdx.y * (T * VEC) + threadIdx.x * VEC;   // 16B aligned
  const int s0 = blockIdx.x * TS;

  // Per-thread taps (W is (D,1,K) -> d*K + k) and bias.
  float w[VEC][Kc];
#pragma unroll
  for (int c = 0; c < VEC; ++c)
#pragma unroll
    for (int k = 0; k < Kc; ++k)
      w[c][k] = Wt[(size_t)(d0 + c) * Kc + k];
  const float4 bv = *reinterpret_cast<const float4*>(bias + d0);

  // Sliding window win[0..6] = xpad[s0+s .. s0+s+6] for current output s.
  float4 win[Kc];
  if (s0 == 0) {
    // Halo from conv_state, layout (B, D, K-1).
#pragma unroll
    for (int j = 0; j < Kc - 1; ++j) {
      float t[VEC];
#pragma unroll
      for (int c = 0; c < VEC; ++c)
        t[c] = cs[((size_t)b * Dc + (d0 + c)) * (Kc - 1) + j];
      win[j] = make_float4(t[0], t[1], t[2], t[3]);
    }
  } else {
#pragma unroll
    for (int j = 0; j < Kc - 1; ++j)
      win[j] = *reinterpret_cast<const float4*>(
          x + ((size_t)b * Sc + (s0 - (Kc - 1) + j)) * Dc + d0);
  }

  const float* xrow = x + ((size_t)b * Sc + s0) * Dc + d0;   // stride Dc per step
  float*       orow = out + ((size_t)b * Sc + s0) * Dc + d0;

  auto body = [&](int s, float4 xi) {
    win[Kc - 1] = xi;
    float4 acc = bv;
#pragma unroll
    for (int k = 0; k < Kc; ++k) {
      acc.x = fmaf(win[k].x, w[0][k], acc.x);
      acc.y = fmaf(win[k].y, w[1][k], acc.y);
      acc.z = fmaf(win[k].z, w[2][k], acc.z);
      acc.w = fmaf(win[k].w, w[3][k], acc.w);
    }
    *reinterpret_cast<float4*>(orow + (size_t)s * Dc) = acc;
#pragma unroll
    for (int j = 0; j < Kc - 1; ++j) win[j] = win[j + 1];
  };

#if CC_ASYNC
  // gfx1250 async DMA pipeline: NBUF GLOBAL_LOAD_ASYNC_TO_LDS_B128 in flight
  // through a RING-slot LDS ring; S_WAIT_ASYNCCNT throttles consumption.
  // Steady loop unrolled by RING so every slot index is a compile-time
  // constant (immediate-offset ds_load_b128, zero ring bookkeeping).
  __shared__ float4 stage[RING][T];
  const int tid = threadIdx.x;
#pragma unroll
  for (int p = 0; p < NBUF; ++p)                 // steps 0..3 -> slots 0..3
    async_cp16(xrow + (size_t)p * Dc, &stage[p][tid]);

  for (int g = 0; g < (TS - NBUF) / RING; ++g) { // 12 groups of 5 steps
    const int sbase = g * RING;                  // multiple of RING
#pragma unroll
    for (int u = 0; u < RING; ++u) {
      // step t = sbase+u lives in slot t % RING == u (sbase % RING == 0)
      __builtin_amdgcn_s_wait_asynccnt(NBUF - 1);   // oldest in-flight done
      asm volatile("" ::: "memory");
      float4 xi = stage[u][tid];
      async_cp16(xrow + (size_t)(sbase + u + NBUF) * Dc,
                 &stage[(u + NBUF) % RING][tid]);   // slot (t+NBUF)%RING
      body(sbase + u, xi);
    }
  }
  __builtin_amdgcn_s_wait_asynccnt(0);              // drain
  asm volatile("" ::: "memory");
#pragma unroll
  for (int u = 0; u < NBUF; ++u) {
    // steps TS-NBUF..TS-1; (TS-NBUF) % RING == 0 -> slots 0..NBUF-1
    float4 xi = stage[u][tid];
    body(TS - NBUF + u, xi);
  }
#else
  for (int s = 0; s < TS; ++s) {
    if (s + 8 < TS)
      __builtin_prefetch(xrow + (size_t)(s + 8) * Dc, 0, 0);  // global_prefetch_b8
    float4 xi = *reinterpret_cast<const float4*>(xrow + (size_t)s * Dc);
    body(s, xi);
  }
#endif
}

// new_conv_state[b, d, j] = x[b, S-6+j, d]   (B, D, K-1) layout.
__global__ __launch_bounds__(T)
void causal_conv1d_state(const float* __restrict__ x, float* __restrict__ ns) {
  const int idx = blockIdx.x * T + threadIdx.x;          // over B*D
  if (idx >= Bc * Dc) return;
  const int b = idx / Dc;
  const int d = idx % Dc;
  const float* xb = x + ((size_t)b * Sc + (Sc - (Kc - 1))) * Dc + d;
  float v[Kc - 1];
#pragma unroll
  for (int j = 0; j < Kc - 1; ++j) v[j] = xb[(size_t)j * Dc];
  float* o = ns + (size_t)idx * (Kc - 1);
#pragma unroll
  for (int j = 0; j < Kc - 1; ++j) o[j] = v[j];
}

extern "C" void kernel_launch(void* const* d_in, const int* in_sizes, int n_in,
                              void* d_out, int out_size, void* d_ws, size_t ws_size,
                              hipStream_t stream) {
  (void)in_sizes; (void)n_in; (void)out_size; (void)d_ws; (void)ws_size;
  const float* x    = (const float*)d_in[0];
  const float* cs   = (const float*)d_in[1];
  const float* Wt   = (const float*)d_in[2];
  const float* bias = (const float*)d_in[3];
  float* out = (float*)d_out;
  float* ns  = out + (size_t)Bc * Sc * Dc;   // second tuple element, flat-concat

  dim3 grid(Sc / TS, Dc / (T * VEC), Bc);    // (64, 2, 4) = 512 blocks
  causal_conv1d_fwd<<<grid, T, 0, stream>>>(x, cs, Wt, bias, out);

  const int nstate = Bc * Dc;                // 8192 threads
  causal_conv1d_state<<<(nstate + T - 1) / T, T, 0, stream>>>(x, ns);
}